// CustomizedLinear_30193620091140
// MI455X (gfx1250) — compile-verified
//
#include <hip/hip_runtime.h>

typedef float v2f __attribute__((ext_vector_type(2)));
typedef float v8f __attribute__((ext_vector_type(8)));

#define F_GENES 2000
#define P_PATH  256
#define EMB     768
#define BATCH   64

#define KT 32                                  // K-chunk
#define MT 32                                  // block rows (in B*P space)
#define NT 128                                 // block cols (in EMB)
#define NCHUNK ((F_GENES + KT - 1) / KT)       // 63 (last chunk 16 valid)

#define A_STRIDE 33                            // 32 + 1 pad (odd -> conflict-free row gathers)
#define B_STRIDE 136                           // 128 + 8 pad (k,k+2 row pair lands on disjoint banks)

__global__ __launch_bounds__(256, 2)
void pathway_linear_wmma(const float* __restrict__ x,     // (64, 2000)
                         const float* __restrict__ W,     // (256, 2000), pre-masked
                         const float* __restrict__ bias,  // (256,)
                         const float* __restrict__ E,     // (2000, 768)
                         float* __restrict__ out)         // (64, 256*768) == (16384, 768)
{
    __shared__ float As[2][MT][A_STRIDE];
    __shared__ float Bs[2][KT][B_STRIDE];

    const int tid  = threadIdx.x;
    const int lane = tid & 31;
    const int wave = tid >> 5;
    const int mi   = wave >> 2;        // 0..1 : which 16-row subtile
    const int ni   = wave & 3;         // 0..3 : subtile cols ni*16 and ni*16+64

    const int m_base = blockIdx.y * MT;        // row in flattened (B*P) space
    const int n_base = blockIdx.x * NT;        // col in EMB
    const int cell   = m_base / P_PATH;        // batch index b, constant per block (256 % 32 == 0)
    const int p_base = m_base % P_PATH;

    // ---- loader decomposition ----
    const int a_row = tid >> 3;                // 0..31
    const int a_qk  = (tid & 7) << 2;          // 0,4,...,28

    v8f acc0 = {0.f,0.f,0.f,0.f,0.f,0.f,0.f,0.f};
    v8f acc1 = {0.f,0.f,0.f,0.f,0.f,0.f,0.f,0.f};

    auto load_chunk = [&](int c, int buf) {
        const int k0 = c * KT;
        // A tile: A[row, kk] = x[b, k0+kk] * W[p_base+row, k0+kk]   (float4 per thread)
        {
            const int d0 = k0 + a_qk;
            const int p  = p_base + a_row;
            if (d0 + 3 < F_GENES) {
                const float4 w4 = *reinterpret_cast<const float4*>(&W[(size_t)p * F_GENES + d0]);
                const float4 x4 = *reinterpret_cast<const float4*>(&x[(size_t)cell * F_GENES + d0]);
                As[buf][a_row][a_qk + 0] = w4.x * x4.x;
                As[buf][a_row][a_qk + 1] = w4.y * x4.y;
                As[buf][a_row][a_qk + 2] = w4.z * x4.z;
                As[buf][a_row][a_qk + 3] = w4.w * x4.w;
            } else {
                #pragma unroll
                for (int j = 0; j < 4; ++j) {
                    const int d = d0 + j;
                    As[buf][a_row][a_qk + j] = (d < F_GENES)
                        ? W[(size_t)p * F_GENES + d] * x[(size_t)cell * F_GENES + d] : 0.0f;
                }
            }
        }
        // B tile: E[k0+kk, n_base+qn .. +3]  (4 float4 per thread, coalesced along n)
        #pragma unroll
        for (int i = 0; i < 4; ++i) {
            const int f4 = tid + i * 256;      // float4 index 0..1023
            const int kk = f4 >> 5;            // 0..31
            const int qn = (f4 & 31) << 2;     // 0..124
            const int d  = k0 + kk;
            if (d < F_GENES) {
                const float4 e4 = *reinterpret_cast<const float4*>(&E[(size_t)d * EMB + n_base + qn]);
                Bs[buf][kk][qn + 0] = e4.x;
                Bs[buf][kk][qn + 1] = e4.y;
                Bs[buf][kk][qn + 2] = e4.z;
                Bs[buf][kk][qn + 3] = e4.w;
            } else {
                Bs[buf][kk][qn + 0] = 0.f;
                Bs[buf][kk][qn + 1] = 0.f;
                Bs[buf][kk][qn + 2] = 0.f;
                Bs[buf][kk][qn + 3] = 0.f;
            }
        }
    };

    load_chunk(0, 0);
    __syncthreads();

    const int row   = (lane & 15) + mi * 16;   // A-frag row (M)
    const int col   = (lane & 15) + ni * 16;   // B-frag col (N)
    const int khalf = (lane >> 4) << 1;        // lanes 0-15 -> K{0,1}; lanes 16-31 -> K{2,3}

    int buf = 0;
    for (int c = 0; c < NCHUNK; ++c) {
        if (c + 1 < NCHUNK) load_chunk(c + 1, buf ^ 1);

        #pragma unroll
        for (int ks = 0; ks < KT / 4; ++ks) {
            const int kp = ks * 4 + khalf;
            v2f a;
            a[0] = As[buf][row][kp];
            a[1] = As[buf][row][kp + 1];
            v2f b0;
            b0[0] = Bs[buf][kp][col];
            b0[1] = Bs[buf][kp + 1][col];
            v2f b1;
            b1[0] = Bs[buf][kp][col + 64];
            b1[1] = Bs[buf][kp + 1][col + 64];
            acc0 = __builtin_amdgcn_wmma_f32_16x16x4_f32(false, a, false, b0,
                                                         (short)0, acc0, false, false);
            acc1 = __builtin_amdgcn_wmma_f32_16x16x4_f32(false, a, false, b1,
                                                         (short)0, acc1, false, false);
        }
        __syncthreads();
        buf ^= 1;
    }

    // Epilogue: C layout (16x16 f32): VGPR i -> M = i (lanes 0-15) / i+8 (lanes 16-31), N = lane%16.
    const int n_loc = lane & 15;
    const int m_add = (lane >> 4) << 3;
    #pragma unroll
    for (int i = 0; i < 8; ++i) {
        const int m = i + m_add;
        const int p = p_base + mi * 16 + m;
        const int r = m_base + mi * 16 + m;
        const float bv = bias[p];
        const size_t o0 = (size_t)r * EMB + n_base + ni * 16 + n_loc;
        out[o0]      = acc0[i] + bv;
        out[o0 + 64] = acc1[i] + bv;
    }
}

extern "C" void kernel_launch(void* const* d_in, const int* in_sizes, int n_in,
                              void* d_out, int out_size, void* d_ws, size_t ws_size,
                              hipStream_t stream) {
    (void)in_sizes; (void)n_in; (void)out_size; (void)d_ws; (void)ws_size;
    const float* x    = (const float*)d_in[0];   // (64, 2000)
    const float* W    = (const float*)d_in[1];   // (256, 2000) already masked
    const float* bias = (const float*)d_in[2];   // (256,)
    // d_in[3] = mask (2000,256): binary, already folded into W -> re-applying is identity; skip.
    const float* E    = (const float*)d_in[4];   // (2000, 768)
    float* out        = (float*)d_out;           // (64, 256*768)

    dim3 grid(EMB / NT, (BATCH * P_PATH) / MT);  // (6, 512)
    dim3 block(256);
    pathway_linear_wmma<<<grid, block, 0, stream>>>(x, W, bias, E, out);
}